// GATv2Conv_19396072309346
// MI455X (gfx1250) — compile-verified
//
#include <hip/hip_runtime.h>
#include <hip/hip_bf16.h>

// ---------------- problem constants (match reference) ----------------
#define Nn   20000
#define Ee   160000
#define INC  128
#define Hh   8
#define Cc   64
#define HC   512          // Hh * Cc
#define NEG_SLOPE 0.2f

typedef float v2f __attribute__((ext_vector_type(2)));
typedef float v8f __attribute__((ext_vector_type(8)));
typedef unsigned int u32x4 __attribute__((ext_vector_type(4)));
typedef int i32x4 __attribute__((ext_vector_type(4)));
typedef int i32x8 __attribute__((ext_vector_type(8)));

// ------- order-preserving float<->uint encoding for atomic segment-max -------
__device__ __forceinline__ unsigned fenc(float f) {
    unsigned u = __float_as_uint(f);
    return (u & 0x80000000u) ? ~u : (u | 0x80000000u);
}
__device__ __forceinline__ float fdec(unsigned u) {
    unsigned v = (u & 0x80000000u) ? (u & 0x7FFFFFFFu) : ~u;
    return __uint_as_float(v);
}

// ---------------------------- small utility kernels ----------------------------
__global__ void gat_zero_f32(float* __restrict__ p, int n) {
    int i = blockIdx.x * 256 + threadIdx.x;
    if (i < n) p[i] = 0.0f;
}

__global__ void gat_init_out(float* __restrict__ out, const float* __restrict__ bias) {
    int i = blockIdx.x * 256 + threadIdx.x;
    if (i < Nn * HC) out[i] = bias[i & (HC - 1)];
}

// count incoming edges per node (segment_sum of ones over dst)
__global__ void gat_edge_count(const int* __restrict__ dst, float* __restrict__ cnt) {
    int e = blockIdx.x * 256 + threadIdx.x;
    if (e < Ee) unsafeAtomicAdd(&cnt[dst[e]], 1.0f);
}

// segment-sum of edge_attr over dst: thread = (edge, 4-float chunk)
__global__ void gat_edge_attr_sum(const int* __restrict__ dst,
                                  const float* __restrict__ ea,
                                  float* __restrict__ sum) {
    int t = blockIdx.x * 256 + threadIdx.x;
    if (t >= Ee * (INC / 4)) return;
    int e  = t >> 5;            // / 32 chunks
    int ch = (t & 31) * 4;
    int d  = dst[e];
    float4 v = *reinterpret_cast<const float4*>(ea + (size_t)e * INC + ch);
    float* s = sum + (size_t)d * INC + ch;
    unsafeAtomicAdd(s + 0, v.x);
    unsafeAtomicAdd(s + 1, v.y);
    unsafeAtomicAdd(s + 2, v.z);
    unsafeAtomicAdd(s + 3, v.w);
}

// loop_attr[n] = sum_attr[n] / max(cnt[n], 1)
__global__ void gat_loop_attr(const float* __restrict__ cnt,
                              const float* __restrict__ sum,
                              float* __restrict__ la) {
    int t = blockIdx.x * 256 + threadIdx.x;
    if (t >= Nn * (INC / 4)) return;
    int n  = t >> 5;
    int ch = (t & 31) * 4;
    float c   = cnt[n];
    float inv = 1.0f / (c > 1.0f ? c : 1.0f);
    float4 v = *reinterpret_cast<const float4*>(sum + (size_t)n * INC + ch);
    float4 r = make_float4(v.x * inv, v.y * inv, v.z * inv, v.w * inv);
    *reinterpret_cast<float4*>(la + (size_t)n * INC + ch) = r;
}

// ---------------------------- TDM helper ----------------------------
// Issue a 2-D TENSOR_LOAD_TO_LDS: tile (tile_d0 x tile_d1) of 4-byte elements
// from a row-major tensor (row stride stride0 elements) into LDS at lds_addr,
// with hardware LDS row padding (pad codes per CDNA5 D# spec §8.4).
// This toolchain exposes the 6-arg builtin:
//   (uint32x4 g0, int32x8 g1, int32x4 g2, int32x4 g3, int32x8 g4, i32 cpol)
__device__ __forceinline__ void tdm_load_2d_f32(unsigned lds_addr, const void* gptr,
                                                unsigned tensor_d0, unsigned tensor_d1,
                                                unsigned tile_d0, unsigned tile_d1,
                                                unsigned long long stride0,
                                                unsigned pad_interval_code,
                                                unsigned pad_amount_code) {
    unsigned long long ga = (unsigned long long)(uintptr_t)gptr;
    u32x4 g0;
    g0.x = 0x1u;                                  // count=1, user-mode descriptor
    g0.y = lds_addr;                              // LDS byte address
    g0.z = (unsigned)(ga & 0xFFFFFFFFull);        // global_addr[31:0]
    g0.w = (unsigned)((ga >> 32) & 0x01FFFFFFull) // global_addr[56:32]
         | (2u << 30);                            // type = 2 ("image")
    i32x8 g1;
    g1[0] = (int)((2u << 16)                      // data_size = 4 bytes
                | (1u << 20)                      // pad_enable
                | (pad_interval_code << 22)
                | (pad_amount_code << 25));
    g1[1] = (int)((tensor_d0 & 0xFFFFu) << 16);   // [15:0] barrier addr=0 | dim0 lo16
    g1[2] = (int)((tensor_d0 >> 16) | ((tensor_d1 & 0xFFFFu) << 16));
    g1[3] = (int)((tensor_d1 >> 16) | (tile_d0 << 16));
    g1[4] = (int)(tile_d1 & 0xFFFFu);             // tile_dim1 | tile_dim2=0 (2-D)
    g1[5] = (int)(unsigned)(stride0 & 0xFFFFFFFFull);
    g1[6] = (int)(unsigned)((stride0 >> 32) & 0xFFFFull);  // stride hi | dim1_stride=0
    g1[7] = 0;
    i32x4 z4 = {0, 0, 0, 0};                      // groups 2/3 unused (<=2-D tensor)
    i32x8 z8 = {0, 0, 0, 0, 0, 0, 0, 0};
    __builtin_amdgcn_tensor_load_to_lds(g0, g1, z4, z4, z8, 0);
}

// ---------------------------- WMMA f32 GEMM with TDM staging ----------------------------
// Y[rows x 512] = X[rows x 128] * W[128 x 512]
// block = 256 threads = 8 wave32 arranged 2(M) x 4(N); block tile 32M x 256N.
// Wave 0 DMA-loads the 32x128 A tile and the whole 128x256 B panel via the
// Tensor Data Mover (hardware-padded LDS rows), waits TENSORcnt, barrier, then
// pure V_WMMA_F32_16X16X4_F32 compute over the full K=128.
#define LDA 132          // A LDS row stride: TDM pad +4 DW per 128-DW row (132 % 64 = 4)
#define LDB 272          // B LDS row stride: TDM pad +16 DW per 256-DW row

__global__ __launch_bounds__(256)
void gat_gemm_f32_wmma(const float* __restrict__ X,
                       const float* __restrict__ W,
                       float* __restrict__ Y) {
    __shared__ float As[32 * LDA];    // 16.9 KB
    __shared__ float Bs[128 * LDB];   // 139.3 KB  (total 156.2 KB LDS)

    const int tid  = threadIdx.x;
    const int lane = tid & 31;
    const int wave = tid >> 5;       // 0..7
    const int wrow = wave >> 2;      // 0..1  -> M offset 16*wrow
    const int wcol = wave & 3;       // 0..3  -> N offset 64*wcol
    const int row0 = blockIdx.x * 32;
    const int col0 = blockIdx.y * 256;

    const int hi   = (lane >= 16) ? 1 : 0;   // lane group (K split for A/B frags)
    const int l16  = lane & 15;
    const int arow = wrow * 16 + l16;        // A fragment row this lane serves

    if (wave == 0) {
        // A tile: 32 rows x 128 cols, row = 128 DW = 64x8B -> interval code 6, +4 DW pad (code 3)
        tdm_load_2d_f32((unsigned)(uintptr_t)&As[0], X + (size_t)row0 * INC,
                        /*tensor_d0=*/INC, /*tensor_d1=*/32,
                        /*tile_d0=*/INC,   /*tile_d1=*/32,
                        /*stride0=*/INC, /*pad_interval=*/6u, /*pad_amount=*/3u);
        // B panel: 128 rows x 256 cols, row = 256 DW = 128x8B -> interval code 7, +16 DW pad (code 15)
        tdm_load_2d_f32((unsigned)(uintptr_t)&Bs[0], W + col0,
                        /*tensor_d0=*/256, /*tensor_d1=*/INC,
                        /*tile_d0=*/256,   /*tile_d1=*/INC,
                        /*stride0=*/HC, /*pad_interval=*/7u, /*pad_amount=*/15u);
        __builtin_amdgcn_s_wait_tensorcnt(0);
    }
    __syncthreads();

    v8f acc[4];
    #pragma unroll
    for (int i = 0; i < 4; ++i)
        #pragma unroll
        for (int j = 0; j < 8; ++j) acc[i][j] = 0.0f;

    #pragma unroll 4
    for (int k = 0; k < INC; k += 4) {
        // A fragment: lanes 0-15 hold K={k,k+1}, lanes 16-31 hold K={k+2,k+3}
        v2f a = *reinterpret_cast<const v2f*>(&As[arow * LDA + (k + 2 * hi)]);
        #pragma unroll
        for (int nt = 0; nt < 4; ++nt) {
            int n = wcol * 64 + nt * 16 + l16;
            v2f b;
            b.x = Bs[(k + 2 * hi    ) * LDB + n];
            b.y = Bs[(k + 2 * hi + 1) * LDB + n];
            acc[nt] = __builtin_amdgcn_wmma_f32_16x16x4_f32(
                /*neg_a=*/false, a, /*neg_b=*/false, b,
                /*c_mod=*/(short)0, acc[nt],
                /*reuse_a=*/false, /*reuse_b=*/false);
        }
    }

    // ---- store: C/D layout — VGPR v: lanes 0-15 => M=v, lanes 16-31 => M=v+8 ----
    const int mbase = row0 + wrow * 16 + hi * 8;
    #pragma unroll
    for (int nt = 0; nt < 4; ++nt) {
        int n = col0 + wcol * 64 + nt * 16 + l16;
        #pragma unroll
        for (int v = 0; v < 8; ++v) {
            Y[(size_t)(mbase + v) * HC + n] = acc[nt][v];
        }
    }
}

// ---------------------------- attention scoring ----------------------------
// thread = (edge-with-loops, head): logit = att_h . leaky_relu(xl[s]+xr[d]+xe[e])
__global__ void gat_score(const float* __restrict__ xl, const float* __restrict__ xr,
                          const float* __restrict__ xe,
                          const int* __restrict__ src, const int* __restrict__ dst,
                          const float* __restrict__ att,
                          float* __restrict__ logits, unsigned* __restrict__ lmaxK) {
    int t = blockIdx.x * 256 + threadIdx.x;
    if (t >= (Ee + Nn) * Hh) return;
    int e = t >> 3, h = t & 7;
    int s, d;
    if (e < Ee) { s = src[e]; d = dst[e]; } else { s = e - Ee; d = s; }

    const float4* pl = reinterpret_cast<const float4*>(xl + (size_t)s * HC + h * Cc);
    const float4* pr = reinterpret_cast<const float4*>(xr + (size_t)d * HC + h * Cc);
    const float4* pe = reinterpret_cast<const float4*>(xe + (size_t)e * HC + h * Cc);
    const float4* pa = reinterpret_cast<const float4*>(att + h * Cc);

    float acc = 0.0f;
    #pragma unroll 4
    for (int i = 0; i < Cc / 4; ++i) {
        float4 a = pl[i], b = pr[i], c = pe[i], w = pa[i];
        float mx = a.x + b.x + c.x;
        float my = a.y + b.y + c.y;
        float mz = a.z + b.z + c.z;
        float mw = a.w + b.w + c.w;
        mx = mx > 0.0f ? mx : NEG_SLOPE * mx;
        my = my > 0.0f ? my : NEG_SLOPE * my;
        mz = mz > 0.0f ? mz : NEG_SLOPE * mz;
        mw = mw > 0.0f ? mw : NEG_SLOPE * mw;
        acc += mx * w.x + my * w.y + mz * w.z + mw * w.w;
    }
    logits[t] = acc;
    atomicMax(&lmaxK[d * Hh + h], fenc(acc));
}

// p = exp(logit - segmax); accumulate denom
__global__ void gat_prob(const int* __restrict__ dst, const unsigned* __restrict__ lmaxK,
                         float* __restrict__ logits, float* __restrict__ denom) {
    int t = blockIdx.x * 256 + threadIdx.x;
    if (t >= (Ee + Nn) * Hh) return;
    int e = t >> 3, h = t & 7;
    int d = (e < Ee) ? dst[e] : (e - Ee);
    float p = __expf(logits[t] - fdec(lmaxK[d * Hh + h]));
    logits[t] = p;
    unsafeAtomicAdd(&denom[d * Hh + h], p);
}

// out[d] += (p/denom) * xl[s]
__global__ void gat_aggregate(const int* __restrict__ src, const int* __restrict__ dst,
                              const float* __restrict__ xl,
                              const float* __restrict__ logits,
                              const float* __restrict__ denom,
                              float* __restrict__ out) {
    int t = blockIdx.x * 256 + threadIdx.x;
    if (t >= (Ee + Nn) * Hh) return;
    int e = t >> 3, h = t & 7;
    int s, d;
    if (e < Ee) { s = src[e]; d = dst[e]; } else { s = e - Ee; d = s; }
    float alpha = logits[t] / denom[d * Hh + h];
    const float4* pl = reinterpret_cast<const float4*>(xl + (size_t)s * HC + h * Cc);
    float* po = out + (size_t)d * HC + h * Cc;
    #pragma unroll 4
    for (int i = 0; i < Cc / 4; ++i) {
        float4 v = pl[i];
        unsafeAtomicAdd(po + 4 * i + 0, alpha * v.x);
        unsafeAtomicAdd(po + 4 * i + 1, alpha * v.y);
        unsafeAtomicAdd(po + 4 * i + 2, alpha * v.z);
        unsafeAtomicAdd(po + 4 * i + 3, alpha * v.w);
    }
}

// ---------------------------- host launcher ----------------------------
extern "C" void kernel_launch(void* const* d_in, const int* in_sizes, int n_in,
                              void* d_out, int out_size, void* d_ws, size_t ws_size,
                              hipStream_t stream) {
    const float* x          = (const float*)d_in[0];
    const int*   edge_index = (const int*)  d_in[1];   // [2, E]
    const float* edge_attr  = (const float*)d_in[2];
    const float* Wl         = (const float*)d_in[3];
    const float* Wr         = (const float*)d_in[4];
    const float* We         = (const float*)d_in[5];
    const float* att        = (const float*)d_in[6];
    const float* bias       = (const float*)d_in[7];
    float* out = (float*)d_out;

    const int* srcp = edge_index;
    const int* dstp = edge_index + Ee;

    // workspace layout (floats)
    float*    wsf       = (float*)d_ws;
    float*    cnt       = wsf;                                   // N
    float*    sumattr   = cnt + Nn;                              // N*128
    float*    denom     = sumattr + (size_t)Nn * INC;            // N*8
    unsigned* lmaxK     = (unsigned*)(denom + (size_t)Nn * Hh);  // N*8
    float*    loop_attr = (float*)(lmaxK + (size_t)Nn * Hh);     // N*128
    float*    xl        = loop_attr + (size_t)Nn * INC;          // N*512
    float*    xr        = xl + (size_t)Nn * HC;                  // N*512
    float*    xe        = xr + (size_t)Nn * HC;                  // (E+N)*512
    float*    logits    = xe + (size_t)(Ee + Nn) * HC;           // (E+N)*8

    dim3 blk(256);

    // 1) zero the accumulation scratch (cnt, sumattr, denom, lmaxK contiguous)
    int nzero = Nn * (1 + INC + Hh + Hh);
    gat_zero_f32<<<(nzero + 255) / 256, blk, 0, stream>>>(wsf, nzero);

    // 2) out = bias (broadcast)
    gat_init_out<<<(Nn * HC + 255) / 256, blk, 0, stream>>>(out, bias);

    // 3) self-loop attr = mean of incoming edge attrs
    gat_edge_count<<<(Ee + 255) / 256, blk, 0, stream>>>(dstp, cnt);
    gat_edge_attr_sum<<<(Ee * (INC / 4) + 255) / 256, blk, 0, stream>>>(dstp, edge_attr, sumattr);
    gat_loop_attr<<<(Nn * (INC / 4) + 255) / 256, blk, 0, stream>>>(cnt, sumattr, loop_attr);

    // 4) projections via f32 WMMA GEMM with TDM staging (row counts divisible by 32)
    gat_gemm_f32_wmma<<<dim3(Nn / 32, 2), blk, 0, stream>>>(x, Wl, xl);
    gat_gemm_f32_wmma<<<dim3(Nn / 32, 2), blk, 0, stream>>>(x, Wr, xr);
    gat_gemm_f32_wmma<<<dim3(Ee / 32, 2), blk, 0, stream>>>(edge_attr, We, xe);
    gat_gemm_f32_wmma<<<dim3(Nn / 32, 2), blk, 0, stream>>>(loop_attr, We, xe + (size_t)Ee * HC);

    // 5) attention logits + segment max
    int nt = (Ee + Nn) * Hh;
    gat_score<<<(nt + 255) / 256, blk, 0, stream>>>(xl, xr, xe, srcp, dstp, att, logits, lmaxK);

    // 6) softmax numerator + denominator
    gat_prob<<<(nt + 255) / 256, blk, 0, stream>>>(dstp, lmaxK, logits, denom);

    // 7) weighted scatter-add aggregation
    gat_aggregate<<<(nt + 255) / 256, blk, 0, stream>>>(srcp, dstp, xl, logits, denom, out);
}